// ATVP_router_UNI_10411000726195
// MI455X (gfx1250) — compile-verified
//
#include <hip/hip_runtime.h>
#include <hip/hip_bf16.h>

// ---------------------------------------------------------------------------
// Problem constants (from reference): B=1024, N=10 groups, G=7 experts,
// K=1024, D=1536.  Output: [B, D] float32.
// ---------------------------------------------------------------------------
#define BATCH 1024
#define NGRP  10
#define NEXP  7
#define KDIM  1024
#define DDIM  1536

typedef __attribute__((ext_vector_type(2))) float v2f;
typedef __attribute__((ext_vector_type(8))) float v8f;

// ---------------------------------------------------------------------------
// 1) Mean over the 7 experts of W_proj ([10,7,K,D] -> Wall slots 0..9) and of
//    W_ib ([7,K,D] -> Wall slot 10).  Pure bandwidth (~490 MB streamed once)
//    -> float4 loads/stores (global_load_b128).
// ---------------------------------------------------------------------------
__global__ __launch_bounds__(256)
void wmean_kernel(const float4* __restrict__ Wp, const float4* __restrict__ Wib,
                  float4* __restrict__ Wall) {
  const long KD4 = (long)KDIM * DDIM / 4;
  long i = (long)blockIdx.x * 256 + threadIdx.x;
  if (i >= 11 * KD4) return;
  int slot = (int)(i / KD4);
  long off = i - (long)slot * KD4;
  float4 s = make_float4(0.f, 0.f, 0.f, 0.f);
  if (slot < NGRP) {
    const float4* p = Wp + ((long)slot * NEXP) * KD4 + off;
#pragma unroll
    for (int e = 0; e < NEXP; ++e) {
      float4 v = p[(long)e * KD4];
      s.x += v.x; s.y += v.y; s.z += v.z; s.w += v.w;
    }
  } else {
#pragma unroll
    for (int e = 0; e < NEXP; ++e) {
      float4 v = Wib[(long)e * KD4 + off];
      s.x += v.x; s.y += v.y; s.z += v.z; s.w += v.w;
    }
  }
  const float r = 1.f / 7.f;
  s.x *= r; s.y *= r; s.z *= r; s.w *= r;
  Wall[i] = s;
}

// Mean of biases: b_proj [10,7,D] -> bmean slots 0..9, b_ib [7,D] -> slot 10.
__global__ __launch_bounds__(256)
void bmean_kernel(const float* __restrict__ bp, const float* __restrict__ bib,
                  float* __restrict__ bmean) {
  int i = blockIdx.x * 256 + threadIdx.x;
  if (i >= 11 * DDIM) return;
  int slot = i / DDIM;
  int d = i - slot * DDIM;
  float s = 0.f;
  if (slot < NGRP) {
#pragma unroll
    for (int e = 0; e < NEXP; ++e) s += bp[((long)slot * NEXP + e) * DDIM + d];
  } else {
#pragma unroll
    for (int e = 0; e < NEXP; ++e) s += bib[(long)e * DDIM + d];
  }
  bmean[i] = s * (1.f / 7.f);
}

// ---------------------------------------------------------------------------
// 2) Router input: xp[b, j*1024+k] = x_enc[{0,4,8}[j]][b][k]   ([B,3072])
// ---------------------------------------------------------------------------
__global__ __launch_bounds__(256)
void build_xp_kernel(const float4* __restrict__ x_enc, float4* __restrict__ xp) {
  const int K4 = KDIM / 4;          // 256
  long i = (long)blockIdx.x * 256 + threadIdx.x;
  const long total = (long)BATCH * 3 * K4;
  if (i >= total) return;
  int b = (int)(i / (3 * K4));
  int c = (int)(i - (long)b * (3 * K4));
  int j = c / K4;                   // 0..2
  int k = c - j * K4;
  int idx = j * 4;                  // 0, 4, 8
  xp[i] = x_enc[((long)idx * BATCH + b) * K4 + k];
}

// ---------------------------------------------------------------------------
// 3) Generic tiled WMMA f32 GEMM:  C[z] = A[z] @ Bm[z] + bias[z]
//    A row-major [M,K], Bm row-major [K,N], C row-major [M,N].
//    Block = 256 threads = 8 waves (2x4), wave tile 32x32 (2x2 wmma frags),
//    block tile 64x128, K staged through LDS (float4 transfers) in steps
//    of 32 -> 32 v_wmma_f32_16x16x4_f32 per stage.
//    Requires: M % 64 == 0, K % 32 == 0, N % 4 == 0 (true for all uses).
// ---------------------------------------------------------------------------
#define GBM 64
#define GBN 128
#define GKT 32
#define APAD 36   // As row stride (dwords): 16 distinct banks for frag reads
#define BPAD 132  // Bs row stride (dwords): de-conflicts the two half-waves

__global__ __launch_bounds__(256)
void gemm_wmma_f32(const float* __restrict__ A, long sAz,
                   const float* __restrict__ Bm, long sBz,
                   const float* __restrict__ bias, long sbz,
                   float* __restrict__ C, long sCz,
                   int M, int N, int K) {
  __shared__ __align__(16) float As[GBM][APAD];   // 9.0 KB
  __shared__ __align__(16) float Bs[GKT][BPAD];   // 16.5 KB

  const int z = blockIdx.z;
  A  += (long)z * sAz;
  Bm += (long)z * sBz;
  C  += (long)z * sCz;
  const float* bz = bias + (long)z * sbz;

  const int tid  = threadIdx.x;
  const int lane = tid & 31;       // wave32
  const int wave = tid >> 5;       // 0..7
  const int wm   = wave >> 2;      // wave row 0..1
  const int wn   = wave & 3;       // wave col 0..3
  const int l16  = lane & 15;
  const int lh   = lane >> 4;      // 0/1

  const int row0 = blockIdx.y * GBM;
  const int col0 = blockIdx.x * GBN;

  v8f acc[2][2] = {};

  for (int ks = 0; ks < K; ks += GKT) {
    // Stage A tile: 64x32 floats = 512 float4, 2 per thread (b128 path).
#pragma unroll
    for (int t = 0; t < (GBM * GKT) / (4 * 256); ++t) {
      int i  = tid + t * 256;
      int m  = i >> 3;             // 0..63
      int k4 = (i & 7) << 2;       // 0,4,...,28
      const float4 v =
          *(const float4*)(A + (long)(row0 + m) * K + (ks + k4));
      *(float4*)&As[m][k4] = v;
    }
    // Stage B tile: 32x128 floats = 1024 float4, 4 per thread, N-guarded.
#pragma unroll
    for (int t = 0; t < (GKT * GBN) / (4 * 256); ++t) {
      int i  = tid + t * 256;
      int k  = i >> 5;             // 0..31
      int n4 = (i & 31) << 2;      // 0,4,...,124
      int gn = col0 + n4;
      float4 v = make_float4(0.f, 0.f, 0.f, 0.f);
      if (gn < N) v = *(const float4*)(Bm + (long)(ks + k) * N + gn);
      *(float4*)&Bs[k][n4] = v;
    }
    __syncthreads();

#pragma unroll
    for (int k4 = 0; k4 < GKT; k4 += 4) {
      v2f af[2], bf[2];
#pragma unroll
      for (int fm = 0; fm < 2; ++fm) {
        int m = wm * 32 + fm * 16 + l16;
        af[fm].x = As[m][k4 + lh * 2 + 0];
        af[fm].y = As[m][k4 + lh * 2 + 1];
      }
#pragma unroll
      for (int fn = 0; fn < 2; ++fn) {
        int n = wn * 32 + fn * 16 + l16;
        bf[fn].x = Bs[k4 + lh * 2 + 0][n];
        bf[fn].y = Bs[k4 + lh * 2 + 1][n];
      }
#pragma unroll
      for (int fm = 0; fm < 2; ++fm)
#pragma unroll
        for (int fn = 0; fn < 2; ++fn)
          acc[fm][fn] = __builtin_amdgcn_wmma_f32_16x16x4_f32(
              false, af[fm], false, bf[fn], (short)0, acc[fm][fn],
              false, false);
    }
    __syncthreads();
  }

  // Epilogue: C/D frag layout -> M = frag_base + 8*lh + v, N = frag_base + l16.
#pragma unroll
  for (int fm = 0; fm < 2; ++fm) {
#pragma unroll
    for (int fn = 0; fn < 2; ++fn) {
      int n = col0 + wn * 32 + fn * 16 + l16;
      if (n < N) {
        float bv = bz[n];
#pragma unroll
        for (int v = 0; v < 8; ++v) {
          int m = row0 + wm * 32 + fm * 16 + lh * 8 + v;
          C[(long)m * N + n] = acc[fm][fn][v] + bv;
        }
      }
    }
  }
}

// ---------------------------------------------------------------------------
// 4) BatchNorm1d (training-mode stats over batch) + activation, per feature.
//    One block per feature column.  act: 0 = relu, 1 = tanh.
// ---------------------------------------------------------------------------
__global__ __launch_bounds__(256)
void bn_act_kernel(const float* __restrict__ X, float* __restrict__ Y,
                   int Bn, int F, int act) {
  const int f = blockIdx.x;
  float s1 = 0.f, s2 = 0.f;
  for (int b = threadIdx.x; b < Bn; b += 256) {
    float x = X[(long)b * F + f];
    s1 += x;
    s2 += x * x;
  }
  __shared__ float r1[256], r2[256];
  r1[threadIdx.x] = s1;
  r2[threadIdx.x] = s2;
  __syncthreads();
  for (int s = 128; s > 0; s >>= 1) {
    if (threadIdx.x < s) {
      r1[threadIdx.x] += r1[threadIdx.x + s];
      r2[threadIdx.x] += r2[threadIdx.x + s];
    }
    __syncthreads();
  }
  float mean = r1[0] / (float)Bn;
  float var  = r2[0] / (float)Bn - mean * mean;
  float inv  = rsqrtf(var + 1e-5f);
  for (int b = threadIdx.x; b < Bn; b += 256) {
    float x = (X[(long)b * F + f] - mean) * inv;
    Y[(long)b * F + f] = act ? tanhf(x) : fmaxf(x, 0.f);
  }
}

// ---------------------------------------------------------------------------
// 5) Final router layer [100 -> 11] + sigmoid, *10, softmax.  One thread / b.
// ---------------------------------------------------------------------------
__global__ __launch_bounds__(256)
void router_final_kernel(const float* __restrict__ h2,
                         const float* __restrict__ w3,
                         const float* __restrict__ b3,
                         float* __restrict__ wgt, int Bn) {
  int b = blockIdx.x * 256 + threadIdx.x;
  if (b >= Bn) return;
  float h[100];
#pragma unroll
  for (int i = 0; i < 100; ++i) h[i] = h2[(long)b * 100 + i];
  float lg[11];
  float mx = -1e30f;
#pragma unroll
  for (int j = 0; j < 11; ++j) {
    float s = b3[j];
    for (int i = 0; i < 100; ++i) s += h[i] * w3[i * 11 + j];
    lg[j] = 10.f / (1.f + __expf(-s));   // sigmoid(s) * 10
    mx = fmaxf(mx, lg[j]);
  }
  float den = 0.f;
#pragma unroll
  for (int j = 0; j < 11; ++j) {
    lg[j] = __expf(lg[j] - mx);
    den += lg[j];
  }
  float r = 1.f / den;
#pragma unroll
  for (int j = 0; j < 11; ++j) wgt[(long)b * 11 + j] = lg[j] * r;
}

// ---------------------------------------------------------------------------
// 6) uni = l2norm(0.1 * (x_ib @ Wib_mean + bib_mean) + 0.9 * x_uni),
//    in-place on expert slot 10.  One block per batch row.
// ---------------------------------------------------------------------------
__global__ __launch_bounds__(256)
void uni_finalize_kernel(float* __restrict__ slot10,
                         const float* __restrict__ x_uni) {
  const int b = blockIdx.x;
  float* row = slot10 + (long)b * DDIM;
  const float* u = x_uni + (long)b * DDIM;
  float vals[DDIM / 256];
  float ss = 0.f;
#pragma unroll
  for (int j = 0; j < DDIM / 256; ++j) {
    int i = threadIdx.x + j * 256;
    float v = 0.1f * row[i] + 0.9f * u[i];
    vals[j] = v;
    ss += v * v;
  }
  __shared__ float red[256];
  red[threadIdx.x] = ss;
  __syncthreads();
  for (int s = 128; s > 0; s >>= 1) {
    if (threadIdx.x < s) red[threadIdx.x] += red[threadIdx.x + s];
    __syncthreads();
  }
  float inv = 1.f / fmaxf(sqrtf(red[0]), 1e-12f);
#pragma unroll
  for (int j = 0; j < DDIM / 256; ++j) {
    int i = threadIdx.x + j * 256;
    row[i] = vals[j] * inv;
  }
}

// ---------------------------------------------------------------------------
// 7) Gated combine of 11 expert rows + final L2 norm.  One block per row.
// ---------------------------------------------------------------------------
__global__ __launch_bounds__(256)
void combine_norm_kernel(const float* __restrict__ groups,
                         const float* __restrict__ wgt,
                         float* __restrict__ out, long slotStride) {
  const int b = blockIdx.x;
  __shared__ float w[11];
  if (threadIdx.x < 11) w[threadIdx.x] = wgt[(long)b * 11 + threadIdx.x];
  __syncthreads();
  float vals[DDIM / 256];
  float ss = 0.f;
#pragma unroll
  for (int j = 0; j < DDIM / 256; ++j) {
    int i = threadIdx.x + j * 256;
    float a = 0.f;
#pragma unroll
    for (int s = 0; s < 11; ++s)
      a += groups[(long)s * slotStride + (long)b * DDIM + i] * w[s];
    vals[j] = a;
    ss += a * a;
  }
  __shared__ float red[256];
  red[threadIdx.x] = ss;
  __syncthreads();
  for (int s = 128; s > 0; s >>= 1) {
    if (threadIdx.x < s) red[threadIdx.x] += red[threadIdx.x + s];
    __syncthreads();
  }
  float inv = 1.f / fmaxf(sqrtf(red[0]), 1e-12f);
#pragma unroll
  for (int j = 0; j < DDIM / 256; ++j) {
    int i = threadIdx.x + j * 256;
    out[(long)b * DDIM + i] = vals[j] * inv;
  }
}

// ---------------------------------------------------------------------------
// Host side
// ---------------------------------------------------------------------------
extern "C" void kernel_launch(void* const* d_in, const int* in_sizes, int n_in,
                              void* d_out, int out_size, void* d_ws, size_t ws_size,
                              hipStream_t stream) {
  (void)in_sizes; (void)n_in; (void)out_size; (void)ws_size;
  const float* x_enc  = (const float*)d_in[0];   // [10,1024,1024]
  const float* x_ib   = (const float*)d_in[1];   // [1024,1024]
  const float* x_uni  = (const float*)d_in[2];   // [1024,1536]
  const float* W_proj = (const float*)d_in[3];   // [10,7,1024,1536]
  const float* b_proj = (const float*)d_in[4];   // [10,7,1536]
  const float* W_ib   = (const float*)d_in[5];   // [7,1024,1536]
  const float* b_ib   = (const float*)d_in[6];   // [7,1536]
  const float* r_w1   = (const float*)d_in[7];   // [3072,512]
  const float* r_b1   = (const float*)d_in[8];   // [512]
  const float* r_w2   = (const float*)d_in[9];   // [512,100]
  const float* r_b2   = (const float*)d_in[10];  // [100]
  const float* r_w3   = (const float*)d_in[11];  // [100,11]
  const float* r_b3   = (const float*)d_in[12];  // [11]
  float* out = (float*)d_out;                    // [1024,1536]

  const long KD = (long)KDIM * DDIM;
  const long BD = (long)BATCH * DDIM;

  float* ws = (float*)d_ws;
  float* Wall   = ws;  ws += 11 * KD;            // mean weights, slots 0..10
  float* bmean  = ws;  ws += 11 * DDIM;          // mean biases
  float* groups = ws;  ws += 11 * BD;            // expert outputs (slot 10 = ib)
  float* xp     = ws;  ws += (long)BATCH * 3072; // router input
  float* hp1    = ws;  ws += (long)BATCH * 512;
  float* h1     = ws;  ws += (long)BATCH * 512;
  float* hp2    = ws;  ws += (long)BATCH * 100;
  float* h2     = ws;  ws += (long)BATCH * 100;
  float* wgt    = ws;  ws += (long)BATCH * 11;

  // 1) fold expert mean into weights / biases (streams W_proj once, b128 path)
  {
    long total4 = 11 * KD / 4;
    wmean_kernel<<<dim3((unsigned)((total4 + 255) / 256)), 256, 0, stream>>>(
        (const float4*)W_proj, (const float4*)W_ib, (float4*)Wall);
    bmean_kernel<<<dim3((11 * DDIM + 255) / 256), 256, 0, stream>>>(
        b_proj, b_ib, bmean);
  }
  // 2) router input (float4 copy)
  {
    long total4 = (long)BATCH * 3072 / 4;
    build_xp_kernel<<<dim3((unsigned)((total4 + 255) / 256)), 256, 0, stream>>>(
        (const float4*)x_enc, (float4*)xp);
  }
  // 3) expert GEMMs: 10 group slots in one batched launch (grid.z), ib separate
  gemm_wmma_f32<<<dim3(DDIM / GBN, BATCH / GBM, NGRP), 256, 0, stream>>>(
      x_enc, (long)BATCH * KDIM, Wall, KD, bmean, DDIM,
      groups, BD, BATCH, DDIM, KDIM);
  gemm_wmma_f32<<<dim3(DDIM / GBN, BATCH / GBM, 1), 256, 0, stream>>>(
      x_ib, 0, Wall + 10 * KD, 0, bmean + 10 * DDIM, 0,
      groups + 10 * BD, 0, BATCH, DDIM, KDIM);
  // 4) router layer 1: [1024,3072]@[3072,512]
  gemm_wmma_f32<<<dim3(512 / GBN, BATCH / GBM, 1), 256, 0, stream>>>(
      xp, 0, r_w1, 0, r_b1, 0, hp1, 0, BATCH, 512, 3072);
  bn_act_kernel<<<512, 256, 0, stream>>>(hp1, h1, BATCH, 512, /*relu*/0);
  // 5) router layer 2: [1024,512]@[512,100]
  gemm_wmma_f32<<<dim3(1, BATCH / GBM, 1), 256, 0, stream>>>(
      h1, 0, r_w2, 0, r_b2, 0, hp2, 0, BATCH, 100, 512);
  bn_act_kernel<<<100, 256, 0, stream>>>(hp2, h2, BATCH, 100, /*tanh*/1);
  // 6) router layer 3 + sigmoid + softmax
  router_final_kernel<<<BATCH / 256, 256, 0, stream>>>(h2, r_w3, r_b3, wgt, BATCH);
  // 7) finalize uni slot (0.1*proj + 0.9*x_uni, l2norm)
  uni_finalize_kernel<<<BATCH, 256, 0, stream>>>(groups + 10 * BD, x_uni);
  // 8) gated combine + final l2norm
  combine_norm_kernel<<<BATCH, 256, 0, stream>>>(groups, wgt, out, BD);
}